// GaussianMixtureUnif_64974265254282
// MI455X (gfx1250) — compile-verified
//
#include <hip/hip_runtime.h>
#include <math.h>

// GMM-EM step, N=200000 rows, C=512 centers, D=3.
// X.mu0^T is a thin-K f32 GEMM -> V_WMMA_F32_16X16X4_F32 (K padded 3->4).
// One wave (32 lanes) per workgroup: WMMA needs EXEC==all-ones, and single-wave
// WGs make LDS staging race-free (lockstep) without atomics.
// Round 2: B operands in LDS + capped unroll to stay under 256 VGPRs
// (round 1 asm showed s_set_vgpr_msb churn from >256-VGPR allocation).

#define C_ 512
#define D_ 3

typedef __attribute__((ext_vector_type(2))) float v2f;
typedef __attribute__((ext_vector_type(4))) float v4f;
typedef __attribute__((ext_vector_type(8))) float v8f;

// workspace layout (float offsets); total ~6.2K floats (~25 KB)
enum {
  WS_SIGMA    = 0,
  WS_GD2      = 1,
  WS_SIGNEW   = 2,
  WS_PI       = 16,            // 512
  WS_LOGPI    = 528,           // 512
  WS_MM       = 1040,          // 512  ||mu0_c||^2
  WS_GSUM     = 1552,          // 512  sum_n gamma
  WS_GX       = 2064,          // 1536 sum_n gamma*x
  WS_MUNEW    = 3600,          // 1536
  WS_MMNEW    = 5136,          // 512
  WS_LOGPINEW = 5648           // 512
};

// ---------------------------------------------------------------- setup ----
__global__ void k_setup(const float* __restrict__ mu0, const float* __restrict__ w,
                        float* __restrict__ ws, float* __restrict__ out, int N) {
  __shared__ float s_max, s_logsum;
  int t = threadIdx.x;
  if (t == 0) {
    float wmax = -INFINITY;
    for (int c = 0; c < C_; ++c) wmax = fmaxf(wmax, w[c]);
    float wsum = 0.f;
    for (int c = 0; c < C_; ++c) wsum += __expf(w[c] - wmax);
    // sigma = 0.1 * sqrt(sum_j var_unbiased(mu0[:,j])) / C^(1/D)
    float vs = 0.f;
    for (int j = 0; j < D_; ++j) {
      float m = 0.f;
      for (int c = 0; c < C_; ++c) m += mu0[c * D_ + j];
      m /= (float)C_;
      float v = 0.f;
      for (int c = 0; c < C_; ++c) { float d = mu0[c * D_ + j] - m; v += d * d; }
      vs += v / (float)(C_ - 1);
    }
    float sigma = 0.1f * sqrtf(vs) / powf((float)C_, 1.0f / (float)D_);
    s_max = wmax; s_logsum = __logf(wsum);
    ws[WS_SIGMA] = sigma;
    ws[WS_GD2]   = 0.f;
    out[(size_t)N * 3] = 0.f;            // Cfe accumulator (re-zero every call)
  }
  __syncthreads();
  int c = t;                              // blockDim.x == 512
  float lp = (w[c] - s_max) - s_logsum;
  ws[WS_LOGPI + c] = lp;
  ws[WS_PI + c]    = __expf(lp);
  float m0 = mu0[c * 3], m1 = mu0[c * 3 + 1], m2 = mu0[c * 3 + 2];
  ws[WS_MM + c] = m0 * m0 + m1 * m1 + m2 * m2;
  ws[WS_GSUM + c] = 0.f;
  ws[WS_GX + 3 * c + 0] = 0.f;
  ws[WS_GX + 3 * c + 1] = 0.f;
  ws[WS_GX + 3 * c + 2] = 0.f;
}

// --------------------------------------------------------------- pass 1 ----
// Per wave: 16-row tile; 32 WMMAs (16 centers each) give dot(x,mu) for the
// whole 16x512 block. gamma tile staged in LDS so we can normalize after the
// per-row sum is known; sum(gamma*D2) folded in per-row (no D2 storage).
__global__ void __launch_bounds__(32) k_pass1(const float* __restrict__ X,
                                              const float* __restrict__ mu0,
                                              float* __restrict__ ws, int N) {
  __shared__ float gld[32][32][8];        // 32 KB  [ctile][lane][k]
  __shared__ v2f   bL[32][32];            //  8 KB  per-lane B operands
  __shared__ float accg[2][C_];           //  4 KB
  __shared__ float accgx[3][2][C_];       // 12 KB
  __shared__ float xrow[16][4];           // x0,x1,x2,||x||^2
  __shared__ float mmL[C_], piL[C_];      //  4 KB

  const int lane = threadIdx.x;
  const int cl = lane & 15;
  const int hi = lane >> 4;

  const float sigma = ws[WS_SIGMA];
  const float inv2s2 = 1.0f / (2.0f * sigma * sigma);

  for (int c = lane; c < C_; c += 32) {
    mmL[c] = ws[WS_MM + c];
    piL[c] = ws[WS_PI + c];
    accg[0][c] = 0.f; accg[1][c] = 0.f;
    for (int j = 0; j < 3; ++j) { accgx[j][0][c] = 0.f; accgx[j][1][c] = 0.f; }
  }
  // B operands: B[k][n] = mu0[cb*16+n][k] (K row 3 padded with 0)
  for (int cb = 0; cb < 32; ++cb) {
    const float* m = mu0 + (cb * 16 + cl) * 3;
    v2f bb;
    bb.x = hi ? m[2] : m[0];
    bb.y = hi ? 0.0f : m[1];
    bL[cb][lane] = bb;
  }
  __syncthreads();

  float gd2_reg = 0.0f;
  const int ntiles = (N + 15) >> 4;
  for (int tile = blockIdx.x; tile < ntiles; tile += gridDim.x) {
    const int row0 = tile * 16;
    __syncthreads();
    if (lane < 16) {
      int r = row0 + lane;
      float x0 = 0.f, x1 = 0.f, x2 = 0.f;
      if (r < N) { x0 = X[r * 3]; x1 = X[r * 3 + 1]; x2 = X[r * 3 + 2]; }
      xrow[lane][0] = x0; xrow[lane][1] = x1; xrow[lane][2] = x2;
      xrow[lane][3] = x0 * x0 + x1 * x1 + x2 * x2;
    }
    __syncthreads();

    // A matrix: lane<16 holds (x0,x1) of row=lane; lane>=16 holds (x2,0)
    v2f a;
    a.x = hi ? xrow[cl][2] : xrow[cl][0];
    a.y = hi ? 0.0f        : xrow[cl][1];

    float xx8[8], xr0[8], xr1[8], xr2[8], valid[8];
#pragma unroll
    for (int k = 0; k < 8; ++k) {
      int rr = k + 8 * hi;                // C/D layout: VGPR k -> row k (+8 for hi half)
      xx8[k] = xrow[rr][3]; xr0[k] = xrow[rr][0];
      xr1[k] = xrow[rr][1]; xr2[k] = xrow[rr][2];
      valid[k] = (row0 + rr < N) ? 1.0f : 0.0f;
    }

    float rs[8] = {0}, rgd2[8] = {0};
#pragma unroll 4
    for (int cb = 0; cb < 32; ++cb) {
      v8f accv = {};
      accv = __builtin_amdgcn_wmma_f32_16x16x4_f32(false, a, false, bL[cb][lane],
                                                   (short)0, accv, false, false);
      const int c = cb * 16 + cl;
      const float mmc = mmL[c], pic = piL[c];
      float garr[8];
#pragma unroll
      for (int k = 0; k < 8; ++k) {
        float d2 = fmaxf(xx8[k] + mmc - 2.0f * accv[k], 0.0f);
        float g  = pic * __expf(-d2 * inv2s2) * valid[k];
        garr[k] = g;
        rs[k]   += g;
        rgd2[k] += g * d2;
      }
      // two b128 stores instead of 8 scalar stores
      *(v4f*)&gld[cb][lane][0] = *(v4f*)&garr[0];
      *(v4f*)&gld[cb][lane][4] = *(v4f*)&garr[4];
    }
    // per-row sums: reduce across the 16 lanes of each half
#pragma unroll
    for (int m = 1; m < 16; m <<= 1) {
#pragma unroll
      for (int k = 0; k < 8; ++k) {
        rs[k]   += __shfl_xor(rs[k],   m, 32);
        rgd2[k] += __shfl_xor(rgd2[k], m, 32);
      }
    }
    float invs[8];
#pragma unroll
    for (int k = 0; k < 8; ++k) invs[k] = (rs[k] > 0.f) ? 1.0f / rs[k] : 0.0f;
    if (cl == 0) {                         // lanes 0 and 16: rows 0-7 / 8-15
      float s = 0.f;
#pragma unroll
      for (int k = 0; k < 8; ++k) s += rgd2[k] * invs[k];
      gd2_reg += s;
    }
    // phase 2: normalize staged gamma, accumulate per-center sums in LDS
#pragma unroll 4
    for (int cb = 0; cb < 32; ++cb) {
      v4f glo = *(const v4f*)&gld[cb][lane][0];
      v4f ghi = *(const v4f*)&gld[cb][lane][4];
      float pg = 0.f, p0 = 0.f, p1 = 0.f, p2 = 0.f;
#pragma unroll
      for (int k = 0; k < 8; ++k) {
        float gv = (k < 4) ? glo[k & 3] : ghi[k & 3];
        float ga = gv * invs[k];
        pg += ga; p0 += ga * xr0[k]; p1 += ga * xr1[k]; p2 += ga * xr2[k];
      }
      const int c = cb * 16 + cl;          // (hi,c) unique per lane: no races
      accg[hi][c]      += pg;
      accgx[0][hi][c]  += p0;
      accgx[1][hi][c]  += p1;
      accgx[2][hi][c]  += p2;
    }
  }
  __syncthreads();
  for (int c = lane; c < C_; c += 32) {
    atomicAdd(&ws[WS_GSUM + c],       accg[0][c] + accg[1][c]);
    atomicAdd(&ws[WS_GX + 3 * c],     accgx[0][0][c] + accgx[0][1][c]);
    atomicAdd(&ws[WS_GX + 3 * c + 1], accgx[1][0][c] + accgx[1][1][c]);
    atomicAdd(&ws[WS_GX + 3 * c + 2], accgx[2][0][c] + accgx[2][1][c]);
  }
  gd2_reg += __shfl_xor(gd2_reg, 16, 32);
  if (lane == 0) atomicAdd(&ws[WS_GD2], gd2_reg);
}

// ------------------------------------------------------------- finalize ----
__global__ void k_finalize(float* __restrict__ ws, int N) {
  int c = threadIdx.x;                    // blockDim.x == 512
  float gs  = ws[WS_GSUM + c];
  float inv = (gs != 0.f) ? 1.0f / gs : 0.0f;
  float m0 = ws[WS_GX + 3 * c] * inv;
  float m1 = ws[WS_GX + 3 * c + 1] * inv;
  float m2 = ws[WS_GX + 3 * c + 2] * inv;
  ws[WS_MUNEW + 3 * c]     = m0;
  ws[WS_MUNEW + 3 * c + 1] = m1;
  ws[WS_MUNEW + 3 * c + 2] = m2;
  ws[WS_MMNEW + c]    = m0 * m0 + m1 * m1 + m2 * m2;
  ws[WS_LOGPINEW + c] = __logf(gs / (float)N);
  if (c == 0) ws[WS_SIGNEW] = sqrtf(ws[WS_GD2] / (3.0f * (float)N));
}

// --------------------------------------------------------------- pass 2 ----
// Recompute gamma (same WMMA tile), accumulate Y = gamma @ mu_new and Cfe.
// Uses log(gamma) = log(pi) - D2/(2s^2) - log(s) so everything accumulates
// unnormalized with g; single phase, no LDS gamma staging needed.
__global__ void __launch_bounds__(32) k_pass2(const float* __restrict__ X,
                                              const float* __restrict__ mu0,
                                              const float* __restrict__ ws,
                                              float* __restrict__ out, int N) {
  __shared__ v2f   bL[32][32];            // 8 KB
  __shared__ float xrow[16][4];
  __shared__ float mmL[C_], piL[C_], ctermL[C_];
  __shared__ float munewL[C_][3];

  const int lane = threadIdx.x;
  const int cl = lane & 15;
  const int hi = lane >> 4;

  const float sigma   = ws[WS_SIGMA];
  const float inv2s2o = 1.0f / (2.0f * sigma * sigma);
  const float sn      = ws[WS_SIGNEW];
  const float inv2s2n = 1.0f / (2.0f * sn * sn);
  const float logsn   = __logf(sn);

  for (int c = lane; c < C_; c += 32) {
    mmL[c] = ws[WS_MM + c];
    piL[c] = ws[WS_PI + c];
    // per-center constant: ||mu_new||^2/(2sn^2) - log(pi_new) + log(pi_old)
    ctermL[c] = ws[WS_MMNEW + c] * inv2s2n - ws[WS_LOGPINEW + c] + ws[WS_LOGPI + c];
    munewL[c][0] = ws[WS_MUNEW + 3 * c];
    munewL[c][1] = ws[WS_MUNEW + 3 * c + 1];
    munewL[c][2] = ws[WS_MUNEW + 3 * c + 2];
  }
  for (int cb = 0; cb < 32; ++cb) {
    const float* m = mu0 + (cb * 16 + cl) * 3;
    v2f bb;
    bb.x = hi ? m[2] : m[0];
    bb.y = hi ? 0.0f : m[1];
    bL[cb][lane] = bb;
  }
  __syncthreads();

  float cfe_reg = 0.0f;
  const int ntiles = (N + 15) >> 4;
  for (int tile = blockIdx.x; tile < ntiles; tile += gridDim.x) {
    const int row0 = tile * 16;
    __syncthreads();
    if (lane < 16) {
      int r = row0 + lane;
      float x0 = 0.f, x1 = 0.f, x2 = 0.f;
      if (r < N) { x0 = X[r * 3]; x1 = X[r * 3 + 1]; x2 = X[r * 3 + 2]; }
      xrow[lane][0] = x0; xrow[lane][1] = x1; xrow[lane][2] = x2;
      xrow[lane][3] = x0 * x0 + x1 * x1 + x2 * x2;
    }
    __syncthreads();

    v2f a;
    a.x = hi ? xrow[cl][2] : xrow[cl][0];
    a.y = hi ? 0.0f        : xrow[cl][1];

    float xx8[8], valid[8];
#pragma unroll
    for (int k = 0; k < 8; ++k) {
      int rr = k + 8 * hi;
      xx8[k] = xrow[rr][3];
      valid[k] = (row0 + rr < N) ? 1.0f : 0.0f;
    }

    float rs[8] = {0}, rT[8] = {0}, rY0[8] = {0}, rY1[8] = {0}, rY2[8] = {0};
#pragma unroll 4
    for (int cb = 0; cb < 32; ++cb) {
      v8f accv = {};
      accv = __builtin_amdgcn_wmma_f32_16x16x4_f32(false, a, false, bL[cb][lane],
                                                   (short)0, accv, false, false);
      const int c = cb * 16 + cl;
      const float mmc = mmL[c], pic = piL[c], ct = ctermL[c];
      const float u0 = munewL[c][0], u1 = munewL[c][1], u2 = munewL[c][2];
#pragma unroll
      for (int k = 0; k < 8; ++k) {
        float d2 = fmaxf(xx8[k] + mmc - 2.0f * accv[k], 0.0f);
        float g  = pic * __expf(-d2 * inv2s2o) * valid[k];
        rs[k]  += g;
        rT[k]  += g * (ct - d2 * inv2s2o);
        rY0[k] += g * u0; rY1[k] += g * u1; rY2[k] += g * u2;
      }
    }
#pragma unroll
    for (int m = 1; m < 16; m <<= 1) {
#pragma unroll
      for (int k = 0; k < 8; ++k) {
        rs[k]  += __shfl_xor(rs[k],  m, 32);
        rT[k]  += __shfl_xor(rT[k],  m, 32);
        rY0[k] += __shfl_xor(rY0[k], m, 32);
        rY1[k] += __shfl_xor(rY1[k], m, 32);
        rY2[k] += __shfl_xor(rY2[k], m, 32);
      }
    }
    // lanes 0-7 write rows 0-7, lanes 16-23 write rows 8-15
    if (cl < 8) {
      const int k = cl;
      const int row = row0 + k + 8 * hi;
      if (row < N) {
        float inv = 1.0f / rs[k];
        float y0 = rY0[k] * inv, y1 = rY1[k] * inv, y2 = rY2[k] * inv;
        out[row * 3]     = y0;
        out[row * 3 + 1] = y1;
        out[row * 3 + 2] = y2;
        float yy = y0 * y0 + y1 * y1 + y2 * y2;
        cfe_reg += rT[k] * inv - __logf(rs[k]) - yy * inv2s2n + 3.0f * logsn;
      }
    }
  }
#pragma unroll
  for (int m = 1; m < 32; m <<= 1) cfe_reg += __shfl_xor(cfe_reg, m, 32);
  if (lane == 0) atomicAdd(&out[(size_t)N * 3], cfe_reg);
}

// ---------------------------------------------------------------- launch ---
extern "C" void kernel_launch(void* const* d_in, const int* in_sizes, int n_in,
                              void* d_out, int out_size, void* d_ws, size_t ws_size,
                              hipStream_t stream) {
  const float* X   = (const float*)d_in[0];   // [N,3]
  const float* mu0 = (const float*)d_in[1];   // [512,3]
  const float* w   = (const float*)d_in[2];   // [512,1]
  float* out = (float*)d_out;                 // [N*3 Y | 1 Cfe]
  float* ws  = (float*)d_ws;
  const int N = in_sizes[0] / 3;

  k_setup<<<1, 512, 0, stream>>>(mu0, w, ws, out, N);
  const int ntiles = (N + 15) / 16;
  const int grid = ntiles < 1024 ? ntiles : 1024;
  k_pass1<<<grid, 32, 0, stream>>>(X, mu0, ws, N);
  k_finalize<<<1, 512, 0, stream>>>(ws, N);
  k_pass2<<<grid, 32, 0, stream>>>(X, mu0, ws, out, N);
}